// SpecNCutout_38062000177638
// MI455X (gfx1250) — compile-verified
//
#include <hip/hip_runtime.h>

// ---------------------------------------------------------------------------
// SpecNCutout for MI455X (gfx1250, wave32).
//   B=128, H=256, W=1024, C=1. out = mask ? mean(x) : x
// Bandwidth-bound 3-kernel pipeline:
//   k_reduce1: read x (128 MiB, RT hint -> stays in 192 MB L2), block partials
//   k_reduce2: fold 1024 partials -> fill = mean(x)
//   k_apply:   re-read x (L2 hit), select fill, NT stores (don't evict x)
// Wave reductions use V_WMMA_F32_16X16X4_F32 (CDNA5 WMMA path) +
// DS_SWIZZLE_B32 SWAPX16 for the cross-half combine.
// ---------------------------------------------------------------------------

#define B_      128
#define H_      256
#define W_      1024
#define NHOLES  5
#define MINW    102   // W // 10
#define MINH    25    // H // 10
#define NTOT    (B_ * H_ * W_)          // 33,554,432 floats
#define NV4     (NTOT / 4)              // 8,388,608 float4
#define R1_BLOCKS  1024
#define R1_THREADS 256
#define R1_ITERS   (NV4 / (R1_BLOCKS * R1_THREADS))   // 32, exact tiling

typedef float v2f __attribute__((ext_vector_type(2)));
typedef float v4f __attribute__((ext_vector_type(4)));
typedef float v8f __attribute__((ext_vector_type(8)));

// Exact wave32 sum-reduction via one f32 WMMA.
// A (16x4): lane L puts its partial at K-slot 0 (lanes 0-15 -> K=0,
// lanes 16-31 -> K=2 per the 32-bit A-matrix layout), zeros elsewhere.
// B (4x16) = ones. Then D[m][n] = s[m] + s[m+16] for every n, so per lane
// sum(D[0..7]) yields sum(t[0..7]) on lanes 0-15 and sum(t[8..15]) on lanes
// 16-31; one SWAPX16 swizzle-add completes the wave total in every lane.
// EXEC must be all 1s (guaranteed: no divergence before this point).
__device__ __forceinline__ float wave_reduce_wmma(float s) {
    v2f a; a[0] = s;    a[1] = 0.0f;
    v2f b; b[0] = 1.0f; b[1] = 1.0f;
    v8f c = {};
    c = __builtin_amdgcn_wmma_f32_16x16x4_f32(
            /*neg_a=*/false, a, /*neg_b=*/false, b,
            /*c_mod=*/(short)0, c, /*reuse_a=*/false, /*reuse_b=*/false);
    float u = ((c[0] + c[1]) + (c[2] + c[3])) + ((c[4] + c[5]) + (c[6] + c[7]));
    // SWAPX16: group-of-32 swizzle, xor_mask=0x10, and_mask=0x1F -> 0x401F
    int ui  = __builtin_bit_cast(int, u);
    int si  = __builtin_amdgcn_ds_swizzle(ui, 0x401F);
    u += __builtin_bit_cast(float, si);
    return u;   // wave total in all lanes
}

// Deterministic 256-thread block reduction (8 waves). Result valid on tid 0.
__device__ __forceinline__ float block_reduce_256(float s) {
    __shared__ float lds[8];
    const float w = wave_reduce_wmma(s);
    const int lane = threadIdx.x & 31;
    const int wid  = threadIdx.x >> 5;
    if (lane == 0) lds[wid] = w;
    __syncthreads();
    float total = 0.0f;
    if (threadIdx.x == 0) {
        #pragma unroll
        for (int i = 0; i < 8; ++i) total += lds[i];
    }
    return total;
}

// Pass 1: per-block partial sums. Exact tiling -> no bounds checks, EXEC full.
// Default (RT) loads intentionally populate L2 so k_apply re-reads hit cache.
__global__ void __launch_bounds__(R1_THREADS)
k_reduce1(const v4f* __restrict__ x4, float* __restrict__ partials) {
    const int tid    = blockIdx.x * R1_THREADS + threadIdx.x;
    const int stride = R1_BLOCKS * R1_THREADS;
    float s = 0.0f;
    #pragma unroll 4
    for (int i = 0; i < R1_ITERS; ++i) {
        const v4f v = x4[tid + i * stride];
        s += (v[0] + v[1]) + (v[2] + v[3]);
    }
    const float t = block_reduce_256(s);
    if (threadIdx.x == 0) partials[blockIdx.x] = t;
}

// Pass 2: fold 1024 partials -> fill value at ws[1024].
__global__ void __launch_bounds__(256)
k_reduce2(float* __restrict__ ws) {
    const int t = threadIdx.x;
    const float s = (ws[t] + ws[t + 256]) + (ws[t + 512] + ws[t + 768]);
    const float total = block_reduce_256(s);
    if (t == 0) ws[1024] = total * (1.0f / (float)NTOT);
}

// Pass 3: one block per (b,h) row; 256 threads x float4 == W.
// Output stores are non-temporal: out is write-once, keep L2 for x.
__global__ void __launch_bounds__(256)
k_apply(const v4f*    __restrict__ x4,
        const int*    __restrict__ xs,
        const int*    __restrict__ ys,
        const int*    __restrict__ xwr,
        const int*    __restrict__ yhr,
        const float*  __restrict__ act,
        const float*  __restrict__ ws,
        v4f*          __restrict__ out4) {
    const int bh = blockIdx.x;          // 0 .. B*H-1
    const int b  = bh >> 8;             // / H_
    const int h  = bh & (H_ - 1);       // % H_
    const float fill = ws[1024];

    // Per-block (uniform) hole rectangles; fully unrolled -> stays in regs.
    int  xlo[NHOLES], xhi[NHOLES];
    bool rowhit[NHOLES];
    #pragma unroll
    for (int n = 0; n < NHOLES; ++n) {
        const int idx = b * NHOLES + n;
        const int xc  = xs[idx];
        const int yc  = ys[idx];
        const int hw  = xwr[idx] + MINW;
        const int hh  = yhr[idx] + MINH;
        int x0 = xc - (hw >> 1); x0 = x0 < 0 ? 0 : (x0 > W_ - 2 ? W_ - 2 : x0);
        int x1 = xc + (hw >> 1); x1 = x1 < 1 ? 1 : (x1 > W_ - 1 ? W_ - 1 : x1);
        int y0 = yc - (hh >> 1); y0 = y0 < 0 ? 0 : (y0 > H_ - 2 ? H_ - 2 : y0);
        int y1 = yc + (hh >> 1); y1 = y1 < 1 ? 1 : (y1 > H_ - 1 ? H_ - 1 : y1);
        const bool a = act[idx] < 1.0f;          // PROB == 1.0
        xlo[n] = x0; xhi[n] = x1;
        rowhit[n] = a && (y0 <= h) && (h <= y1);
    }

    const int w0   = threadIdx.x << 2;
    const int base = (bh << 8) + threadIdx.x;    // float4 index: bh*W/4 + tid
    v4f v = x4[base];
    #pragma unroll
    for (int e = 0; e < 4; ++e) {
        const int w = w0 + e;
        bool m = false;
        #pragma unroll
        for (int n = 0; n < NHOLES; ++n)
            m = m || (rowhit[n] && (xlo[n] <= w) && (w <= xhi[n]));
        v[e] = m ? fill : v[e];
    }
    __builtin_nontemporal_store(v, &out4[base]);
}

extern "C" void kernel_launch(void* const* d_in, const int* in_sizes, int n_in,
                              void* d_out, int out_size, void* d_ws, size_t ws_size,
                              hipStream_t stream) {
    (void)in_sizes; (void)n_in; (void)out_size; (void)ws_size;
    const v4f*    x4  = (const v4f*)  d_in[0];
    const int*    xs  = (const int*)  d_in[1];
    const int*    ys  = (const int*)  d_in[2];
    const int*    xwr = (const int*)  d_in[3];
    const int*    yhr = (const int*)  d_in[4];
    const float*  act = (const float*)d_in[5];
    float*        ws  = (float*)d_ws;     // [0..1023] partials, [1024] fill

    k_reduce1<<<R1_BLOCKS, R1_THREADS, 0, stream>>>(x4, ws);
    k_reduce2<<<1, 256, 0, stream>>>(ws);
    k_apply<<<B_ * H_, 256, 0, stream>>>(x4, xs, ys, xwr, yhr, act, ws,
                                         (v4f*)d_out);
}